// SoftSplat_36009005809817
// MI455X (gfx1250) — compile-verified
//
#include <hip/hip_runtime.h>

// Problem constants (from reference): B=4, C=32, H=512, W=960
constexpr int Bn  = 4;
constexpr int Cn  = 32;
constexpr int Hn  = 512;
constexpr int Wn  = 960;
constexpr int HWn = Hn * Wn;                        // 491,520
constexpr int NPIX = Bn * HWn;                      // 1,966,080 source pixels
constexpr long NOUT = (long)Bn * Cn * HWn;          // 62,914,560 output elems
constexpr int SCAT_BLOCKS_PER_B = HWn / 256;        // 1920 (exact)
constexpr int NORM_BLOCKS_PER_B = (Cn * HWn / 4) / 256; // 15360 (exact)
constexpr int NORM_BLOCKS_PER_PLANE = HWn / 1024;   // 480 (exact)

// Native 128-bit vector type (clang ext_vector) — accepted by the
// __builtin_nontemporal_* builtins, unlike HIP's float4 class.
typedef float v4f __attribute__((ext_vector_type(4)));

// Native CDNA5 non-returning fp32 global atomic add, SADDR addressing mode:
//   global_atomic_add_f32 v_offset32, v_data, s[base:base+1]
// Base pointer must be wave-uniform (we compute it from blockIdx only, so it
// lives in an SGPR pair) -> no 64-bit per-lane address math in the hot loop.
// Non-returning (TH[0]=0): tracked with STOREcnt, no latency exposed.
__device__ __forceinline__ void global_fadd_s(float* sbase, int voff_bytes, float v) {
  asm volatile("global_atomic_add_f32 %0, %1, %2"
               : : "v"(voff_bytes), "v"(v), "s"(sbase));
}

__global__ __launch_bounds__(256) void softsplat_scatter(
    const float* __restrict__ inp,     // (B, C, H, W)
    const float* __restrict__ flow,    // (B, 2, H, W)
    const float* __restrict__ metric,  // (B, 1, H, W)
    const float* __restrict__ beta,    // (1,)
    float* __restrict__ out,           // (B, C, H, W) numerator accumulator
    float* __restrict__ den)           // (B, H, W)    denominator accumulator
{
  // Batch index is uniform per block (HWn % 256 == 0): SALU-only.
  const int b    = blockIdx.x / SCAT_BLOCKS_PER_B;
  const int rblk = blockIdx.x - b * SCAT_BLOCKS_PER_B;
  const int hw   = rblk * 256 + threadIdx.x;       // pixel within (H,W) plane
  const int h    = hw / Wn;
  const int w    = hw - h * Wn;

  // Read-once inputs: non-temporal so they don't evict hot L2 accumulator lines.
  const float* fb = flow + (size_t)b * 2 * HWn;     // uniform base (SGPR)
  const float fdx = __builtin_nontemporal_load(fb + hw);
  const float fdy = __builtin_nontemporal_load(fb + HWn + hw);
  const float m   = __builtin_nontemporal_load(metric + (size_t)b * HWn + hw);
  const float em  = __expf(beta[0] * m);            // v_exp_f32 path

  const float ox  = (float)w + fdx;
  const float oy  = (float)h + fdy;
  const float x0f = floorf(ox), y0f = floorf(oy);
  const int   x0  = (int)x0f,   y0  = (int)y0f;
  const float fx  = ox - x0f,   fy  = oy - y0f;
  const float gx  = 1.0f - fx,  gy  = 1.0f - fy;
  const int   x1  = x0 + 1,     y1  = y0 + 1;

  const bool vx0 = (unsigned)x0 < (unsigned)Wn;
  const bool vx1 = (unsigned)x1 < (unsigned)Wn;
  const bool vy0 = (unsigned)y0 < (unsigned)Hn;
  const bool vy1 = (unsigned)y1 < (unsigned)Hn;

  // Branchless boundary handling: out-of-range corner -> weight 0 (adding 0.0
  // is a no-op and keeps untouched den entries exactly 0.0).
  const float w00 = (vx0 && vy0) ? gx * gy * em : 0.0f;
  const float w10 = (vx1 && vy0) ? fx * gy * em : 0.0f;
  const float w01 = (vx0 && vy1) ? gx * fy * em : 0.0f;
  const float w11 = (vx1 && vy1) ? fx * fy * em : 0.0f;

  const int cx0 = min(max(x0, 0), Wn - 1), cx1 = min(max(x1, 0), Wn - 1);
  const int cy0 = min(max(y0, 0), Hn - 1), cy1 = min(max(y1, 0), Hn - 1);
  // Byte offsets within one (H,W) plane — computed once, reused for all 33
  // scatter targets (den + 32 channels).
  const int o00 = (cy0 * Wn + cx0) * 4, o10 = (cy0 * Wn + cx1) * 4;
  const int o01 = (cy1 * Wn + cx0) * 4, o11 = (cy1 * Wn + cx1) * 4;

  // Denominator splat (channel "32" of the packed tensor in the reference).
  float* db = den + (size_t)b * HWn;                // uniform base (SGPR pair)
  global_fadd_s(db, o00, w00);
  global_fadd_s(db, o10, w10);
  global_fadd_s(db, o01, w01);
  global_fadd_s(db, o11, w11);

  // 32 channels: per-wave coalesced 128B loads per channel, adjacent lanes
  // scatter to adjacent addresses (conflict-free in the L2 atomic units).
  // Per channel: 1 NT load + 4 v_mul + 4 saddr atomics + 2 SALU base bumps.
  const float* ib = inp + (size_t)b * Cn * HWn;     // uniform base
  float*       ob = out + (size_t)b * Cn * HWn;     // uniform base
#pragma unroll
  for (int c = 0; c < Cn; ++c) {
    const float v  = __builtin_nontemporal_load(ib + (size_t)c * HWn + hw);
    float*      oc = ob + (size_t)c * HWn;          // uniform (SGPR) per channel
    global_fadd_s(oc, o00, v * w00);
    global_fadd_s(oc, o10, v * w10);
    global_fadd_s(oc, o01, v * w01);
    global_fadd_s(oc, o11, v * w11);
  }
}

__global__ __launch_bounds__(256) void softsplat_norm(
    float* __restrict__ out, const float* __restrict__ den)
{
  // All index math scalar except the threadIdx term (grids divide exactly).
  const int b    = blockIdx.x / NORM_BLOCKS_PER_B;
  const int rblk = blockIdx.x - b * NORM_BLOCKS_PER_B;
  const int pblk = rblk % NORM_BLOCKS_PER_PLANE;     // block within (H,W) plane
  const int hw   = pblk * 1024 + threadIdx.x * 4;    // multiple of 4

  const v4f d4 = *(const v4f*)(den + (size_t)b * HWn + hw);
  float* op = out + (size_t)blockIdx.x * 1024 + threadIdx.x * 4;
  v4f n4 = *(const v4f*)op;

  n4.x /= (d4.x == 0.0f) ? 1.0f : d4.x;
  n4.y /= (d4.y == 0.0f) ? 1.0f : d4.y;
  n4.z /= (d4.z == 0.0f) ? 1.0f : d4.z;
  n4.w /= (d4.w == 0.0f) ? 1.0f : d4.w;

  // Write-once result: non-temporal 128-bit store.
  __builtin_nontemporal_store(n4, (v4f*)op);
}

extern "C" void kernel_launch(void* const* d_in, const int* in_sizes, int n_in,
                              void* d_out, int out_size, void* d_ws, size_t ws_size,
                              hipStream_t stream) {
  const float* inp    = (const float*)d_in[0];  // (B,C,H,W) f32
  const float* flow   = (const float*)d_in[1];  // (B,2,H,W) f32
  const float* metric = (const float*)d_in[2];  // (B,1,H,W) f32
  const float* beta   = (const float*)d_in[3];  // (1,)      f32
  float* out = (float*)d_out;
  float* den = (float*)d_ws;                    // B*H*W floats = 7.86 MB scratch

  // Zero accumulators every call (graph-capturable memset nodes).
  (void)hipMemsetAsync(out, 0, (size_t)out_size * sizeof(float), stream);
  (void)hipMemsetAsync(den, 0, (size_t)Bn * HWn * sizeof(float), stream);

  const dim3 blk(256);
  softsplat_scatter<<<dim3(NPIX / 256), blk, 0, stream>>>(
      inp, flow, metric, beta, out, den);

  softsplat_norm<<<dim3((unsigned)(NOUT / 4 / 256)), blk, 0, stream>>>(out, den);
}